// CLOCModel_71889162600823
// MI455X (gfx1250) — compile-verified
//
#include <hip/hip_runtime.h>
#include <hip/hip_bf16.h>

typedef float v2f __attribute__((ext_vector_type(2)));
typedef float v8f __attribute__((ext_vector_type(8)));
typedef unsigned int u32x4 __attribute__((ext_vector_type(4)));
typedef int i32x4 __attribute__((ext_vector_type(4)));
typedef int i32x8 __attribute__((ext_vector_type(8)));

#define T_STEPS 65536
#define LCH     64            // chunk length
#define NCHUNK  1024          // T_STEPS / LCH
#define DSTATE  512           // 128 nat + 128 unnat + 256 opsin
#define KGH     544           // DSTATE + 32 (input dim)

// Feature probes (evaluate to 0 on the host pass -> fallback paths compile).
#if __has_builtin(__builtin_amdgcn_tensor_load_to_lds) && __has_builtin(__builtin_amdgcn_s_wait_tensorcnt)
#define HAS_TDM 1
#else
#define HAS_TDM 0
#endif
#if __has_builtin(__builtin_amdgcn_global_load_async_to_lds_b32) && __has_builtin(__builtin_amdgcn_s_wait_asynccnt)
#define HAS_ASYNC 1
#else
#define HAS_ASYNC 0
#endif

#define AS_GLOBAL_INT(p) ((__attribute__((address_space(1))) int*)(unsigned long long)(p))
#define AS_LOCAL_INT(p)  ((__attribute__((address_space(3))) int*)(unsigned long long)(p))

// D = A(16x4) * B(4x16) + C, f32 WMMA (CDNA5 V_WMMA_F32_16X16X4_F32)
__device__ __forceinline__ v8f wmma4(v2f a, v2f b, v8f c) {
  return __builtin_amdgcn_wmma_f32_16x16x4_f32(
      /*neg_a=*/false, a, /*neg_b=*/false, b,
      /*c_mod=*/(short)0, c, /*reuse_a=*/false, /*reuse_b=*/false);
}

#if HAS_TDM
// One TDM descriptor: gather a 32x16 f32 tile (16 rows of 32 elements,
// row stride 2048 elements) from global U into LDS at lds_off.
// D# layout per CDNA5 ISA ch.8: group0 = {count, lds_addr, gaddr, type=2},
// group1 = {data_size=4B, tensor_dim0=32, tensor_dim1=16, tile 32x16, stride0=2048}.
__device__ __forceinline__ void tdm_load_u_tile(const float* gsrc, unsigned int lds_off) {
  unsigned long long ga = (unsigned long long)gsrc;
  u32x4 g0;
  g0.x = 1u;                                              // count=1, is_restore=0
  g0.y = lds_off;                                         // lds_addr (bytes)
  g0.z = (unsigned int)ga;                                // global_addr[31:0]
  g0.w = (unsigned int)((ga >> 32) & 0x01FFFFFFu) | (2u << 30);  // addr[56:32] | type=2
  i32x8 g1;
  g1[0] = 0x00020000;                                     // data_size = 2 (4 bytes)
  g1[1] = (int)(32u << 16);                               // tensor_dim0 = 32
  g1[2] = (int)(16u << 16);                               // tensor_dim1 = 16
  g1[3] = (int)(32u << 16);                               // tile_dim0 = 32
  g1[4] = 16;                                             // tile_dim1 = 16
  g1[5] = 2048;                                           // tensor_dim0_stride = 64*32
  g1[6] = 0;
  g1[7] = 0;
  i32x4 z4 = {};
#if __clang_major__ >= 23
  i32x8 z8 = {};
  __builtin_amdgcn_tensor_load_to_lds(g0, g1, z4, z4, z8, 0);
#else
  __builtin_amdgcn_tensor_load_to_lds(g0, g1, z4, z4, 0);
#endif
}
#endif

// ---------------------------------------------------------------------------
// Build combined transition [G | H] (512 x 544, row-major).
// ---------------------------------------------------------------------------
__global__ void build_gh_kernel(float* __restrict__ GH,
                                const float* __restrict__ A_natnat,
                                const float* __restrict__ K_nat,
                                const float* __restrict__ C_y_nat,
                                const float* __restrict__ A_unnat,
                                const float* __restrict__ K_unnat,
                                const float* __restrict__ C_y_unnat,
                                const float* __restrict__ Bp_nat,
                                const float* __restrict__ Bp_unnat,
                                const float* __restrict__ A_opsin,
                                const float* __restrict__ B_opsin,
                                const float* __restrict__ C_opsin) {
  int idx = blockIdx.x * blockDim.x + threadIdx.x;
  if (idx >= DSTATE * KGH) return;
  int row = idx / KGH, col = idx % KGH;
  float v = 0.0f;
  if (row < 128) {
    int i = row;
    if (col < 128)       v = A_natnat[i * 128 + col] + K_nat[i] * C_y_nat[col];
    else if (col < 256)  v = K_nat[i] * C_y_unnat[col - 128];
    else if (col < 512) {
      int j = col - 256; float s = 0.0f;
      for (int m = 0; m < 64; ++m) s += Bp_nat[i * 64 + m] * C_opsin[m * 256 + j];
      v = s;
    }
  } else if (row < 256) {
    int i = row - 128;
    if (col >= 128 && col < 256) {
      int j = col - 128;
      v = A_unnat[i * 128 + j] + K_unnat[i] * C_y_unnat[j];
    } else if (col >= 256 && col < 512) {
      int j = col - 256; float s = 0.0f;
      for (int m = 0; m < 64; ++m) s += Bp_unnat[i * 64 + m] * C_opsin[m * 256 + j];
      v = s;
    }
  } else {
    int i = row - 256;
    if (col >= 256 && col < 512) v = A_opsin[i * 256 + (col - 256)];
    else if (col >= 512)         v = B_opsin[i * 32 + (col - 512)];
  }
  GH[idx] = v;
}

// ---------------------------------------------------------------------------
// Generic f32 WMMA GEMM: C(MxN) = A(MxK)*B(KxN), row-major, one 16x16 tile
// per wave (EXEC all-ones around WMMA), K in steps of 4.
// ---------------------------------------------------------------------------
__global__ __launch_bounds__(256)
void wmma_gemm_kernel(float* __restrict__ C, const float* __restrict__ A,
                      const float* __restrict__ B,
                      int M, int N, int K, int lda, int ldb, int ldc) {
  int lane = threadIdx.x & 31;
  int wave = threadIdx.x >> 5;
  int tilesN = N >> 4, tilesM = M >> 4;
  int tile = blockIdx.x * (blockDim.x >> 5) + wave;
  if (tile >= tilesM * tilesN) return;
  int tm = tile / tilesN, tn = tile - tm * tilesN;
  int half = lane >> 4, l15 = lane & 15;
  int arow = tm * 16 + l15;
  int bcol = tn * 16 + l15;
  __builtin_prefetch(A + (size_t)arow * lda, 0, 1);   // global_prefetch_b8
  v8f acc = {};
  for (int k = 0; k < K; k += 4) {
    int ka = k + half * 2;                            // even -> 8B-aligned A pair
    v2f a = *(const v2f*)(A + (size_t)arow * lda + ka);
    v2f b;
    b.x = B[(size_t)(ka + 0) * ldb + bcol];
    b.y = B[(size_t)(ka + 1) * ldb + bcol];
    acc = wmma4(a, b, acc);
  }
#pragma unroll
  for (int r = 0; r < 8; ++r) {
    int row = tm * 16 + r + half * 8;                 // D layout: VGPR r -> M=r / M=r+8
    C[(size_t)row * ldc + bcol] = acc[r];
  }
}

// Copy H (= G^0 H) into impulse block 0 of F (512 x 2048, block j = G^j H).
__global__ void copy_h_kernel(const float* __restrict__ GH, float* __restrict__ F) {
  int i = blockIdx.x * blockDim.x + threadIdx.x;
  if (i >= DSTATE * 32) return;
  int row = i >> 5, e = i & 31;
  F[(size_t)row * 2048 + e] = GH[row * KGH + 512 + e];
}

// Qt[k][i] = Q[i][k] so the sequential scan reads coalesced columns.
__global__ void transpose512_kernel(const float* __restrict__ Q, float* __restrict__ Qt) {
  int i = blockIdx.x * blockDim.x + threadIdx.x;
  if (i >= DSTATE * DSTATE) return;
  int row = i >> 9, col = i & 511;
  Qt[(size_t)col * DSTATE + row] = Q[i];
}

// ---------------------------------------------------------------------------
// Drive GEMM: V(512 x 1024) = F(512 x 2048) * Ubar, with
// Ubar[32*j + e, c] = U[(c*64 + 63 - j)*32 + e] indexed on the fly.
// ---------------------------------------------------------------------------
__global__ __launch_bounds__(256)
void drive_gemm_kernel(float* __restrict__ V, const float* __restrict__ F,
                       const float* __restrict__ U) {
  int lane = threadIdx.x & 31;
  int wave = threadIdx.x >> 5;
  int tile = blockIdx.x * 8 + wave;                   // 32 x 64 = 2048 tiles
  int tm = tile >> 6, tn = tile & 63;
  int half = lane >> 4, l15 = lane & 15;
  int arow = tm * 16 + l15;
  int bcol = tn * 16 + l15;                           // chunk index
  __builtin_prefetch(F + (size_t)arow * 2048, 0, 1);
  v8f acc = {};
  for (int k = 0; k < 2048; k += 4) {
    int ka = k + half * 2;                            // even; ka,ka+1 share block j
    v2f a = *(const v2f*)(F + (size_t)arow * 2048 + ka);
    int j = ka >> 5, e = ka & 31;
    size_t ub = ((size_t)bcol * LCH + (LCH - 1 - j)) * 32;
    v2f b; b.x = U[ub + e]; b.y = U[ub + e + 1];
    acc = wmma4(a, b, acc);
  }
#pragma unroll
  for (int r = 0; r < 8; ++r) {
    int row = tm * 16 + r + half * 8;
    V[(size_t)row * NCHUNK + bcol] = acc[r];
  }
}

// ---------------------------------------------------------------------------
// Sequential chunk-level scan (only remaining latency chain, 1024 steps):
//   s_{c+1} = G^64 * s_c + v_c ; store all chunk-start states S[c].
// ---------------------------------------------------------------------------
__global__ __launch_bounds__(512)
void chunk_scan_kernel(const float* __restrict__ Qt, const float* __restrict__ V,
                       float* __restrict__ S,
                       const float* __restrict__ xn0, const float* __restrict__ xu0,
                       const float* __restrict__ xo0) {
  __shared__ float s[DSTATE];
  int i = threadIdx.x;
  float w0 = (i < 128) ? xn0[i] : (i < 256) ? xu0[i - 128] : xo0[i - 256];
  s[i] = w0;
  S[i] = w0;
  __syncthreads();
  for (int c = 0; c < NCHUNK; ++c) {
    float r = V[(size_t)i * NCHUNK + c];
    for (int k = 0; k < DSTATE; ++k)
      r = fmaf(Qt[(size_t)k * DSTATE + i], s[k], r);  // s[k] is an LDS broadcast
    __syncthreads();
    s[i] = r;
    S[(size_t)(c + 1) * DSTATE + i] = r;
    __syncthreads();
  }
}

// History row 0 = initial state.
__global__ void init_row0_kernel(const float* __restrict__ xn0,
                                 const float* __restrict__ xu0,
                                 const float* __restrict__ xo0,
                                 float* __restrict__ natH, float* __restrict__ unnatH,
                                 float* __restrict__ opsH) {
  int i = blockIdx.x * blockDim.x + threadIdx.x;
  if (i < 128)       natH[i]         = xn0[i];
  else if (i < 256)  unnatH[i - 128] = xu0[i - 128];
  else if (i < 512)  opsH[i - 256]   = xo0[i - 256];
}

// ---------------------------------------------------------------------------
// Parallel expansion: each WG advances 16 independent chunks as the 16
// columns of X (512x16 in LDS). Each timestep: X_new = [G|H]*[X;u_t] via
// 32 M-tiles of V_WMMA_F32_16X16X4_F32 (K=544), B-fragments from LDS.
// u_{t+1} is DMA'd by the Tensor Data Mover into a double buffer so the
// TDM overlaps WMMA; chunk-start states are staged with async LDS loads.
// ---------------------------------------------------------------------------
__global__ __launch_bounds__(256)
void expand_kernel(const float* __restrict__ GH, const float* __restrict__ S,
                   const float* __restrict__ U,
                   float* __restrict__ natH, float* __restrict__ unnatH,
                   float* __restrict__ opsH) {
  __shared__ float X[DSTATE * 16];                    // state, [row][chunk]
  __shared__ float Ustage[2][32 * 16];                // u_t staging, [chunk][elec]
  int tid = threadIdx.x;
  int lane = tid & 31, wave = tid >> 5;
  int half = lane >> 4, l15 = lane & 15;
  int c0 = blockIdx.x * 16;

  // Stage the 16 chunk-start states (transposing gather).
#if HAS_ASYNC
  for (int i = tid; i < DSTATE * 16; i += 256) {
    int row = i >> 4, n = i & 15;
    const float* g = S + (size_t)(c0 + n) * DSTATE + row;
    __builtin_amdgcn_global_load_async_to_lds_b32(
        AS_GLOBAL_INT(g), AS_LOCAL_INT(&X[i]), 0, 0);
  }
  __builtin_amdgcn_s_wait_asynccnt(0);                // per-wave ASYNCcnt drain
#else
  for (int i = tid; i < DSTATE * 16; i += 256) {
    int row = i >> 4, n = i & 15;
    X[row * 16 + n] = S[(size_t)(c0 + n) * DSTATE + row];
  }
#endif
  __builtin_prefetch(GH + (size_t)(wave * 4) * 16 * KGH, 0, 1);
  __syncthreads();

#if HAS_TDM
  if (wave == 0) {                                    // prime u_0 via TDM
    tdm_load_u_tile(U + (size_t)c0 * LCH * 32,
                    (unsigned int)(unsigned long long)&Ustage[0][0]);
    __builtin_amdgcn_s_wait_tensorcnt(0);
  }
  __syncthreads();
#endif

  for (int t = 0; t < LCH; ++t) {
    int buf = t & 1;
#if HAS_TDM
    if (wave == 0 && t + 1 < LCH)                     // DMA u_{t+1} behind compute
      tdm_load_u_tile(U + ((size_t)c0 * LCH + (t + 1)) * 32,
                      (unsigned int)(unsigned long long)&Ustage[buf ^ 1][0]);
#else
    for (int i = tid; i < 32 * 16; i += 256) {
      int n = i >> 5, e = i & 31;
      Ustage[buf][n * 32 + e] = U[((size_t)(c0 + n) * LCH + t) * 32 + e];
    }
    __syncthreads();
#endif

    v8f acc[4];
    v8f z = {};
#pragma unroll
    for (int q = 0; q < 4; ++q) acc[q] = z;

    // K = 0..511: B-fragments from the state block of LDS.
    for (int k = 0; k < DSTATE; k += 4) {
      int kk = k + half * 2;
      v2f b;
      b.x = X[(kk + 0) * 16 + l15];                   // ds_load
      b.y = X[(kk + 1) * 16 + l15];
#pragma unroll
      for (int q = 0; q < 4; ++q) {                   // 4 M-tiles per wave
        int arow = (wave * 4 + q) * 16 + l15;
        v2f a = *(const v2f*)(GH + (size_t)arow * KGH + kk);
        acc[q] = wmma4(a, b, acc[q]);
      }
    }
    // K = 512..543: B-fragments from the TDM-staged u_t tile.
    for (int k = DSTATE; k < KGH; k += 4) {
      int kk = k + half * 2;
      int e = kk - DSTATE;
      v2f b;
      b.x = Ustage[buf][l15 * 32 + e];
      b.y = Ustage[buf][l15 * 32 + e + 1];
#pragma unroll
      for (int q = 0; q < 4; ++q) {
        int arow = (wave * 4 + q) * 16 + l15;
        v2f a = *(const v2f*)(GH + (size_t)arow * KGH + kk);
        acc[q] = wmma4(a, b, acc[q]);
      }
    }
    __syncthreads();

    // Scatter D to LDS (next-step state) and the global histories.
#pragma unroll
    for (int q = 0; q < 4; ++q) {
#pragma unroll
      for (int r = 0; r < 8; ++r) {
        int row = (wave * 4 + q) * 16 + r + half * 8;
        int n = l15;
        float v = acc[q][r];
        X[row * 16 + n] = v;
        size_t trow = (size_t)(c0 + n) * LCH + t + 1;
        if (row < 128)       natH[trow * 128 + row] = v;
        else if (row < 256)  unnatH[trow * 128 + (row - 128)] = v;
        else                 opsH[trow * 256 + (row - 256)] = v;
      }
    }
#if HAS_TDM
    if (wave == 0) __builtin_amdgcn_s_wait_tensorcnt(0);  // u_{t+1} landed
#endif
    __syncthreads();
  }
}

// y_t = C_y_nat . x_nat_t + C_y_unnat . x_unnat_t, one wave per timestep.
__global__ __launch_bounds__(256)
void y_kernel(const float* __restrict__ Cyn, const float* __restrict__ Cyu,
              const float* __restrict__ natH, const float* __restrict__ unnatH,
              float* __restrict__ y) {
  int w = (blockIdx.x * blockDim.x + threadIdx.x) >> 5;
  int lane = threadIdx.x & 31;
  if (w >= T_STEPS) return;
  const float* xn = natH + (size_t)w * 128;
  const float* xu = unnatH + (size_t)w * 128;
  float acc = 0.0f;
  for (int i = lane; i < 128; i += 32)
    acc = fmaf(Cyn[i], xn[i], fmaf(Cyu[i], xu[i], acc));
  for (int off = 16; off > 0; off >>= 1)
    acc += __shfl_down(acc, off, 32);
  if (lane == 0) y[w] = acc;
}

extern "C" void kernel_launch(void* const* d_in, const int* in_sizes, int n_in,
                              void* d_out, int out_size, void* d_ws, size_t ws_size,
                              hipStream_t stream) {
  (void)in_sizes; (void)n_in; (void)out_size; (void)ws_size;
  const float* xn0      = (const float*)d_in[0];
  const float* xu0      = (const float*)d_in[1];
  const float* xo0      = (const float*)d_in[2];
  const float* U        = (const float*)d_in[3];
  const float* A_natnat = (const float*)d_in[4];
  const float* K_nat    = (const float*)d_in[5];
  const float* C_y_nat  = (const float*)d_in[6];
  const float* A_unnat  = (const float*)d_in[7];
  const float* K_unnat  = (const float*)d_in[8];
  const float* C_y_unnat= (const float*)d_in[9];
  const float* Bp_nat   = (const float*)d_in[10];
  const float* Bp_unnat = (const float*)d_in[11];
  const float* A_opsin  = (const float*)d_in[12];
  const float* B_opsin  = (const float*)d_in[13];
  const float* C_opsin  = (const float*)d_in[14];

  float* out    = (float*)d_out;
  float* y      = out;                                  // T
  float* natH   = y + T_STEPS;                          // (T+1) x 128
  float* unnatH = natH + (size_t)(T_STEPS + 1) * 128;   // (T+1) x 128
  float* opsH   = unnatH + (size_t)(T_STEPS + 1) * 128; // (T+1) x 256

  float* ws = (float*)d_ws;
  float* GH = ws;                    ws += DSTATE * KGH;        // 512 x 544
  float* Q0 = ws;                    ws += DSTATE * DSTATE;     // squaring ping
  float* Q1 = ws;                    ws += DSTATE * DSTATE;     // squaring pong
  float* Qt = ws;                    ws += DSTATE * DSTATE;     // (G^64)^T
  float* F  = ws;                    ws += DSTATE * 2048;       // [H | GH | ... | G^63 H]
  float* V  = ws;                    ws += DSTATE * NCHUNK;     // per-chunk drives
  float* S  = ws;                    ws += (NCHUNK + 1) * DSTATE; // chunk-start states

  // 1) Build [G|H].
  build_gh_kernel<<<(DSTATE * KGH + 255) / 256, 256, 0, stream>>>(
      GH, A_natnat, K_nat, C_y_nat, A_unnat, K_unnat, C_y_unnat,
      Bp_nat, Bp_unnat, A_opsin, B_opsin, C_opsin);

  // 2) G^64 via six WMMA squarings (ping-pong Q0/Q1), then transpose.
  wmma_gemm_kernel<<<128, 256, 0, stream>>>(Q0, GH, GH, 512, 512, 512, KGH, KGH, 512); // G^2
  wmma_gemm_kernel<<<128, 256, 0, stream>>>(Q1, Q0, Q0, 512, 512, 512, 512, 512, 512); // G^4
  wmma_gemm_kernel<<<128, 256, 0, stream>>>(Q0, Q1, Q1, 512, 512, 512, 512, 512, 512); // G^8
  wmma_gemm_kernel<<<128, 256, 0, stream>>>(Q1, Q0, Q0, 512, 512, 512, 512, 512, 512); // G^16
  wmma_gemm_kernel<<<128, 256, 0, stream>>>(Q0, Q1, Q1, 512, 512, 512, 512, 512, 512); // G^32
  wmma_gemm_kernel<<<128, 256, 0, stream>>>(Q1, Q0, Q0, 512, 512, 512, 512, 512, 512); // G^64
  transpose512_kernel<<<1024, 256, 0, stream>>>(Q1, Qt);

  // 3) Impulse response blocks F_j = G^j H.
  copy_h_kernel<<<64, 256, 0, stream>>>(GH, F);
  for (int j = 1; j < LCH; ++j)
    wmma_gemm_kernel<<<8, 256, 0, stream>>>(F + j * 32, GH, F + (j - 1) * 32,
                                            512, 32, 512, KGH, 2048, 2048);

  // 4) All per-chunk drive vectors in one big WMMA GEMM (512 x 2048 x 1024).
  drive_gemm_kernel<<<256, 256, 0, stream>>>(V, F, U);

  // 5) Sequential combine of 1024 chunk summaries.
  chunk_scan_kernel<<<1, 512, 0, stream>>>(Qt, V, S, xn0, xu0, xo0);

  // 6) History row 0 + parallel within-chunk expansion (WMMA + TDM).
  init_row0_kernel<<<2, 256, 0, stream>>>(xn0, xu0, xo0, natH, unnatH, opsH);
  expand_kernel<<<NCHUNK / 16, 256, 0, stream>>>(GH, S, U, natH, unnatH, opsH);

  // 7) y_t from stored histories.
  y_kernel<<<T_STEPS / 8, 256, 0, stream>>>(C_y_nat, C_y_unnat, natH, unnatH, y);
}